// GAT_24489903522141
// MI455X (gfx1250) — compile-verified
//
#include <hip/hip_runtime.h>
#include <math.h>

#define NODES 50000
#define EDGES 800000

typedef __attribute__((ext_vector_type(2))) float v2f;
typedef __attribute__((ext_vector_type(8))) float v8f;

__device__ __forceinline__ void atomic_max_f32(float* addr, float v) {
    // classic signed-max / unsigned-min idiom; valid with -inf init
    if (v >= 0.0f) atomicMax((int*)addr, __float_as_int(v));
    else           atomicMin((unsigned int*)addr, __float_as_uint(v));
}

// ---------------------------------------------------------------------------
// ft[n, c] = sum_d x[n,d] * W[c/64, d, c%64] + bias[c]
// 16-node row tile in LDS (padded), one 16x16 WMMA tile per wave.
// ---------------------------------------------------------------------------
template<int D, int HK>
__global__ __launch_bounds__(32 * (HK / 16))
void gemm_ft_kernel(const float* __restrict__ x, const float* __restrict__ W,
                    const float* __restrict__ bias, float* __restrict__ ft)
{
    constexpr int K   = 64;
    constexpr int NW  = HK / 16;
    constexpr int LDA = D + 1;              // pad to avoid 16-way bank conflicts
    __shared__ float xs[16 * LDA];

    const int tid  = threadIdx.x;
    const int lane = tid & 31;
    const int wave = tid >> 5;
    const int n0   = blockIdx.x << 4;

    for (int i = tid; i < 16 * D; i += 32 * NW) {
        const int r = i / D;
        const int c = i - r * D;
        xs[r * LDA + c] = x[(size_t)(n0 + r) * D + c];
    }
    __syncthreads();

    const int m     = lane & 15;            // A-matrix row (node within tile)
    const int khalf = (lane >> 4) << 1;     // lanes16-31 hold K+2,K+3
    const int col   = (wave << 4) + m;      // output column (head-concat)
    const int h     = col / K;
    const int kk    = col - h * K;
    const float* Wc = W + (size_t)h * D * K + kk;   // stride K per d

    v8f acc = {};
#pragma unroll 8
    for (int d0 = 0; d0 < D; d0 += 4) {
        v2f a, b;
        a.x = xs[m * LDA + d0 + khalf];
        a.y = xs[m * LDA + d0 + khalf + 1];
        b.x = Wc[(size_t)(d0 + khalf)     * K];
        b.y = Wc[(size_t)(d0 + khalf + 1) * K];
        acc = __builtin_amdgcn_wmma_f32_16x16x4_f32(false, a, false, b,
                                                    (short)0, acc, false, false);
    }

    const float bc = bias[col];
    const int   r0 = n0 + ((lane >> 4) << 3);   // C/D: lanes16-31 hold M = j+8
#pragma unroll
    for (int j = 0; j < 8; ++j)
        ft[(size_t)(r0 + j) * HK + col] = acc[j] + bc;
}

// a1[n,h] = ft[n,h,:] . al[h,:] + bl[h]  ;  a2 likewise with ar/br
__global__ void attn_coef_kernel(const float* __restrict__ ft,
                                 const float* __restrict__ al, const float* __restrict__ bl,
                                 const float* __restrict__ ar, const float* __restrict__ br,
                                 float* __restrict__ a1, float* __restrict__ a2, int H)
{
    int i = blockIdx.x * blockDim.x + threadIdx.x;      // i = n*H + h
    if (i >= NODES * H) return;
    int h = i % H;
    const float4* f = (const float4*)(ft + (size_t)i * 64);   // (n*H+h)*64, 16B aligned
    const float4* A = (const float4*)(al + h * 64);
    const float4* B = (const float4*)(ar + h * 64);
    float s1 = 0.f, s2 = 0.f;
#pragma unroll
    for (int k = 0; k < 16; ++k) {
        float4 v = f[k], a = A[k], b = B[k];
        s1 += v.x * a.x + v.y * a.y + v.z * a.z + v.w * a.w;
        s2 += v.x * b.x + v.y * b.y + v.z * b.z + v.w * b.w;
    }
    a1[i] = s1 + bl[h];
    a2[i] = s2 + br[h];
}

__global__ void layer_init_kernel(float* __restrict__ mmax, float* __restrict__ denom,
                                  float4* __restrict__ accum4, int NH, long long NHK4)
{
    long long i = (long long)blockIdx.x * blockDim.x + threadIdx.x;
    if (i < NHK4) accum4[i] = make_float4(0.f, 0.f, 0.f, 0.f);
    if (i < NH) { mmax[i] = -__builtin_inff(); denom[i] = 0.0f; }
}

// per edge: s[e,h] = leaky_relu(a1[dst]+a2[src]) ; segment-max into mmax[dst,h]
__global__ void edge_score_kernel(const int* __restrict__ src, const int* __restrict__ dst,
                                  const float* __restrict__ a1, const float* __restrict__ a2,
                                  float* __restrict__ s, float* __restrict__ mmax, int H)
{
    int e = blockIdx.x * blockDim.x + threadIdx.x;
    if (e >= EDGES) return;
    const int sn = src[e], dn = dst[e];
    for (int h = 0; h < H; ++h) {
        float v = a1[dn * H + h] + a2[sn * H + h];
        v = v > 0.f ? v : 0.01f * v;
        s[e * H + h] = v;
        atomic_max_f32(&mmax[dn * H + h], v);
    }
}

// per edge: s[e,h] <- exp(s - m[dst]) ; segment-sum into denom
__global__ void edge_exp_kernel(const int* __restrict__ dst, float* __restrict__ s,
                                const float* __restrict__ mmax, float* __restrict__ denom, int H)
{
    int e = blockIdx.x * blockDim.x + threadIdx.x;
    if (e >= EDGES) return;
    const int dn = dst[e];
    for (int h = 0; h < H; ++h) {
        float ev = expf(s[e * H + h] - mmax[dn * H + h]);
        s[e * H + h] = ev;
        atomicAdd(&denom[dn * H + h], ev);
    }
}

// accum[dst, 4c..4c+3] += ev[e, h] * ft[src, 4c..4c+3]  (unnormalized)
__global__ void edge_aggregate_kernel(const int* __restrict__ src, const int* __restrict__ dst,
                                      const float* __restrict__ ev, const float* __restrict__ ft,
                                      float* __restrict__ accum, int H, int hkShift)
{
    const int q = hkShift - 2;                       // log2(HK/4)
    long long i = (long long)blockIdx.x * blockDim.x + threadIdx.x;
    if (i >= ((long long)EDGES << q)) return;
    const int c4 = (int)(i & ((1 << q) - 1));        // float4 index within row
    const int e  = (int)(i >> q);
    const int h  = (c4 >> 4) & (H - 1);              // (c4*4)/64, 0 when H==1
    const float w = ev[e * H + h];
    const float4 f = *((const float4*)(ft + ((size_t)src[e] << hkShift)) + c4);
    float* ac = accum + ((size_t)dst[e] << hkShift) + (c4 << 2);
    atomicAdd(ac + 0, w * f.x);
    atomicAdd(ac + 1, w * f.y);
    atomicAdd(ac + 2, w * f.z);
    atomicAdd(ac + 3, w * f.w);
}

// x_out = elu(accum / denom), in place (accum == x_out), float4-wide
__global__ void node_epilogue_kernel(float* __restrict__ accum, const float* __restrict__ denom,
                                     int H, int hkShift)
{
    const int q = hkShift - 2;
    long long i = (long long)blockIdx.x * blockDim.x + threadIdx.x;
    if (i >= ((long long)NODES << q)) return;
    const int c4 = (int)(i & ((1 << q) - 1));
    const int n  = (int)(i >> q);
    const int h  = (c4 >> 4) & (H - 1);
    const float d = denom[n * H + h];
    const float inv = (d > 0.f) ? 1.0f / d : 0.0f;
    float4* ap = (float4*)(accum + ((size_t)n << hkShift)) + c4;
    float4 v = *ap;
    v.x *= inv; v.y *= inv; v.z *= inv; v.w *= inv;
    v.x = v.x > 0.f ? v.x : expm1f(v.x);
    v.y = v.y > 0.f ? v.y : expm1f(v.y);
    v.z = v.z > 0.f ? v.z : expm1f(v.z);
    v.w = v.w > 0.f ? v.w : expm1f(v.w);
    *ap = v;
}

extern "C" void kernel_launch(void* const* d_in, const int* in_sizes, int n_in,
                              void* d_out, int out_size, void* d_ws, size_t ws_size,
                              hipStream_t stream)
{
    const float* features = (const float*)d_in[0];
    const int*   src      = (const int*)d_in[1];
    const int*   dst      = (const int*)d_in[2];
    const float* W0  = (const float*)d_in[3];
    const float* b0  = (const float*)d_in[4];
    const float* al0 = (const float*)d_in[5];
    const float* bl0 = (const float*)d_in[6];
    const float* ar0 = (const float*)d_in[7];
    const float* br0 = (const float*)d_in[8];
    const float* W1  = (const float*)d_in[9];
    const float* b1  = (const float*)d_in[10];
    const float* al1 = (const float*)d_in[11];
    const float* bl1 = (const float*)d_in[12];
    const float* ar1 = (const float*)d_in[13];
    const float* br1 = (const float*)d_in[14];
    const float* Wf  = (const float*)d_in[15];
    const float* bf  = (const float*)d_in[16];
    const float* alf = (const float*)d_in[17];
    const float* blf = (const float*)d_in[18];
    const float* arf = (const float*)d_in[19];
    const float* brf = (const float*)d_in[20];
    float* out = (float*)d_out;
    (void)in_sizes; (void)n_in; (void)out_size; (void)ws_size;

    float* p = (float*)d_ws;
    float* ft    = p; p += (size_t)NODES * 128;
    float* bufA  = p; p += (size_t)NODES * 128;
    float* bufB  = p; p += (size_t)NODES * 128;
    float* a1    = p; p += (size_t)NODES * 2;
    float* a2    = p; p += (size_t)NODES * 2;
    float* mmax  = p; p += (size_t)NODES * 2;
    float* denom = p; p += (size_t)NODES * 2;
    float* ev    = p; p += (size_t)EDGES * 2;

    auto run_edges = [&](const float* ftL, const float* al, const float* bl,
                         const float* ar, const float* br, float* accum, int H) {
        const int HK = H * 64;
        const int hkShift = (HK == 128) ? 7 : 6;
        attn_coef_kernel<<<(NODES * H + 255) / 256, 256, 0, stream>>>(ftL, al, bl, ar, br, a1, a2, H);
        layer_init_kernel<<<(unsigned)(((long long)NODES * (HK / 4) + 255) / 256), 256, 0, stream>>>(
            mmax, denom, (float4*)accum, NODES * H, (long long)NODES * (HK / 4));
        edge_score_kernel<<<(EDGES + 255) / 256, 256, 0, stream>>>(src, dst, a1, a2, ev, mmax, H);
        edge_exp_kernel<<<(EDGES + 255) / 256, 256, 0, stream>>>(dst, ev, mmax, denom, H);
        edge_aggregate_kernel<<<(unsigned)(((long long)EDGES * (HK / 4) + 255) / 256), 256, 0, stream>>>(
            src, dst, ev, ftL, accum, H, hkShift);
        node_epilogue_kernel<<<(unsigned)(((long long)NODES * (HK / 4) + 255) / 256), 256, 0, stream>>>(
            accum, denom, H, hkShift);
    };

    // Layer 0: [N,256] -> ft [N,128] (2 heads x 64)
    gemm_ft_kernel<256, 128><<<NODES / 16, 256, 0, stream>>>(features, W0, b0, ft);
    run_edges(ft, al0, bl0, ar0, br0, bufA, 2);

    // Layer 1: [N,128] -> ft [N,128]
    gemm_ft_kernel<128, 128><<<NODES / 16, 256, 0, stream>>>(bufA, W1, b1, ft);
    run_edges(ft, al1, bl1, ar1, br1, bufB, 2);

    // Final layer: [N,128] -> ft [N,64], 1 head, output straight into d_out
    gemm_ft_kernel<128, 64><<<NODES / 16, 128, 0, stream>>>(bufB, Wf, bf, ft);
    run_edges(ft, alf, blf, arf, brf, out, 1);
}